// EmbeddingGemmaPooler_30657476559490
// MI455X (gfx1250) — compile-verified
//
#include <hip/hip_runtime.h>
#include <hip/hip_bf16.h>
#include <math.h>

typedef __attribute__((ext_vector_type(2))) float v2f;
typedef __attribute__((ext_vector_type(8))) float v8f;

// ---------------------------------------------------------------------------
// Kernel 0: zero the pooled-sum accumulator (must re-init every launch).
// ---------------------------------------------------------------------------
__global__ void zero_kernel(float* __restrict__ p, int n) {
    int i = blockIdx.x * blockDim.x + threadIdx.x;
    if (i < n) p[i] = 0.0f;
}

// ---------------------------------------------------------------------------
// Kernel 1: segment-mean pooling.  grid = (B, NSPLIT), block = H/4 threads.
// Each thread owns one float4 column slot of the row; coalesced b128 loads
// stream the full 192 MiB of hidden_states.  Partial sums are pre-scaled by
// 1/len and merged with global_atomic_add_f32.
// ---------------------------------------------------------------------------
__global__ void pool_kernel(const float* __restrict__ hs,
                            const int* __restrict__ lens,
                            float* __restrict__ pooled, int H) {
    const int seg    = blockIdx.x;
    const int split  = blockIdx.y;
    const int nsplit = gridDim.y;

    // segment start = prefix sum of lengths (uniform scalar loop, B=64)
    long start = 0;
    for (int i = 0; i < seg; ++i) start += (long)lens[i];
    const int len = lens[seg];

    const int rows_per = (len + nsplit - 1) / nsplit;
    const int r0 = split * rows_per;
    const int r1 = (r0 + rows_per < len) ? (r0 + rows_per) : len;
    if (r0 >= r1) return;

    const int h4 = H >> 2;               // float4 per row
    const int c4 = threadIdx.x;          // 0 .. h4-1
    const float4* __restrict__ base = (const float4*)hs;

    float4 acc = make_float4(0.f, 0.f, 0.f, 0.f);
    for (long r = start + r0; r < start + r1; ++r) {
        float4 v = base[r * h4 + c4];
        acc.x += v.x; acc.y += v.y; acc.z += v.z; acc.w += v.w;
    }
    const float inv = 1.0f / (float)len;
    float* dst = pooled + (size_t)seg * H + (size_t)c4 * 4;
    atomicAdd(dst + 0, acc.x * inv);
    atomicAdd(dst + 1, acc.y * inv);
    atomicAdd(dst + 2, acc.z * inv);
    atomicAdd(dst + 3, acc.w * inv);
}

// ---------------------------------------------------------------------------
// Kernel 2: fp32 NT GEMM via V_WMMA_F32_16X16X4_F32.
//   D[M,N] = A[M,K] * Bt[N,K]^T      (A, Bt, D row-major)
// One wave computes one 16x16 output tile, K consumed 4 at a time.
// A fragment (16x4, 2 VGPR):  lane l(0..15): K=0,1 ; lane 16+l: K=2,3 ; M=l.
// B fragment (4x16, 2 VGPR):  symmetric, N=l, same K split.
// D (8 VGPR): vgpr r -> M = r + 8*(lane>=16), N = lane&15.
// block = 128 threads = 4 waves, each wave takes one N tile.
// ---------------------------------------------------------------------------
__global__ void gemm_nt_wmma_f32(const float* __restrict__ A,
                                 const float* __restrict__ Bt,
                                 float* __restrict__ D,
                                 int N, int K) {
    const int lane = threadIdx.x & 31;
    const int wave = threadIdx.x >> 5;
    const int nt   = blockIdx.x * 4 + wave;   // N tile index
    const int mt   = blockIdx.y;              // M tile index
    const int half = lane >> 4;               // 0: K=0,1   1: K=2,3
    const int l    = lane & 15;

    const float* __restrict__ arow = A  + (size_t)(mt * 16 + l) * K + 2 * half;
    const float* __restrict__ brow = Bt + (size_t)(nt * 16 + l) * K + 2 * half;

    v8f c = {0.f, 0.f, 0.f, 0.f, 0.f, 0.f, 0.f, 0.f};
#pragma unroll 4
    for (int k = 0; k < K; k += 4) {
        v2f a = *(const v2f*)(arow + k);
        v2f b = *(const v2f*)(brow + k);
        // 8 args: (neg_a, A, neg_b, B, c_mod, C, reuse_a, reuse_b)
        c = __builtin_amdgcn_wmma_f32_16x16x4_f32(false, a, false, b,
                                                  (short)0, c, false, false);
    }

    float* __restrict__ drow = D + (size_t)(mt * 16 + 8 * half) * N + nt * 16 + l;
#pragma unroll
    for (int r = 0; r < 8; ++r) drow[(size_t)r * N] = c[r];
}

// ---------------------------------------------------------------------------
// Kernel 3: in-place row-wise L2 normalize, torch-style x / max(||x||, eps).
// One block per row (64 rows x 768 cols), wave-shuffle + LDS reduction.
// ---------------------------------------------------------------------------
__global__ void l2norm_kernel(float* __restrict__ y, int H) {
    const int row = blockIdx.x;
    float* __restrict__ p = y + (size_t)row * H;

    __shared__ float red[32];
    float ss = 0.0f;
    for (int c = threadIdx.x; c < H; c += blockDim.x) {
        float v = p[c];
        ss += v * v;
    }
    // wave32 reduction
    for (int off = 16; off > 0; off >>= 1)
        ss += __shfl_down(ss, off, 32);
    const int wave = threadIdx.x >> 5;
    const int lane = threadIdx.x & 31;
    if (lane == 0) red[wave] = ss;
    __syncthreads();
    if (threadIdx.x == 0) {
        float t = 0.0f;
        const int nw = blockDim.x >> 5;
        for (int w = 0; w < nw; ++w) t += red[w];
        red[0] = 1.0f / fmaxf(sqrtf(t), 1e-12f);
    }
    __syncthreads();
    const float s = red[0];
    for (int c = threadIdx.x; c < H; c += blockDim.x) p[c] *= s;
}

// ---------------------------------------------------------------------------
// Launch: zero -> pool -> GEMM1 (pooled x W1^T) -> GEMM2 (h x W2^T) -> l2norm
// ---------------------------------------------------------------------------
extern "C" void kernel_launch(void* const* d_in, const int* in_sizes, int n_in,
                              void* d_out, int out_size, void* d_ws, size_t ws_size,
                              hipStream_t stream) {
    const float* hs   = (const float*)d_in[0];   // [T, H]
    const int*   lens = (const int*)  d_in[1];   // [B]
    const float* W1   = (const float*)d_in[2];   // [FF, H] row-major
    const float* W2   = (const float*)d_in[3];   // [H, FF] row-major
    float*       out  = (float*)d_out;           // [B, H]

    const int B  = in_sizes[1];
    const int H  = out_size / B;                 // 768
    const int FF = in_sizes[2] / H;              // 3072

    float* pooled = (float*)d_ws;                // [B, H]
    float* hbuf   = pooled + (size_t)B * H;      // [B, FF]

    // 0) zero pooled accumulator
    {
        int n = B * H;
        zero_kernel<<<(n + 255) / 256, 256, 0, stream>>>(pooled, n);
    }
    // 1) segment mean pooling (bandwidth-critical: 192 MiB stream)
    {
        const int NSPLIT = 16;
        dim3 grid(B, NSPLIT);
        pool_kernel<<<grid, H / 4, 0, stream>>>(hs, lens, pooled, H);
    }
    // 2) h = pooled @ W1^T   (M=B, N=FF, K=H)
    {
        dim3 grid(FF / 64, B / 16);
        gemm_nt_wmma_f32<<<grid, 128, 0, stream>>>(pooled, W1, hbuf, FF, H);
    }
    // 3) y = h @ W2^T        (M=B, N=H, K=FF)
    {
        dim3 grid(H / 64, B / 16);
        gemm_nt_wmma_f32<<<grid, 128, 0, stream>>>(hbuf, W2, out, H, FF);
    }
    // 4) row-wise L2 normalize in place
    l2norm_kernel<<<B, 256, 0, stream>>>(out, H);
}